// GNN_Network_Norm_Mesh_Enhanced_14181982011841
// MI455X (gfx1250) — compile-verified
//
#include <hip/hip_runtime.h>
#include <hip/hip_bf16.h>

typedef __attribute__((ext_vector_type(2))) float v2f;
typedef __attribute__((ext_vector_type(8))) float v8f;

#define NB 10          // graphs per batch
#define KMAX 48        // max padded K for fc1/fc2
#define ASTR 49        // LDS row stride for A tile (49 coprime with 64 banks)

__device__ __forceinline__ float preluf(float x, float a) { return x >= 0.0f ? x : a * x; }

// ---------------------------------------------------------------- utilities
__global__ void fill_zero(float* __restrict__ p, int n) {
    int i = blockIdx.x * blockDim.x + threadIdx.x;
    if (i < n) p[i] = 0.0f;
}

// ---------------------------------------------------------------- encoders: x->xe (3->30->10), mask->me
__global__ void encoder_kernel(const float* __restrict__ x, const float* __restrict__ mask,
                               const float* __restrict__ ew1, const float* __restrict__ eb1,
                               const float* __restrict__ ew2, const float* __restrict__ eb2,
                               const float* __restrict__ ea1, const float* __restrict__ ea2,
                               const float* __restrict__ mw1, const float* __restrict__ mb1,
                               const float* __restrict__ mw2, const float* __restrict__ mb2,
                               const float* __restrict__ ma1, const float* __restrict__ ma2,
                               float* __restrict__ xe, float* __restrict__ me, int n) {
    // layout per half: [0,90) w1, [90,120) b1, [120,420) w2, [420,430) b2
    __shared__ float sW[860];
    int tid = threadIdx.x;
    for (int i = tid; i < 430; i += blockDim.x) {
        float vx, vm;
        if (i < 90)       { vx = ew1[i];       vm = mw1[i]; }
        else if (i < 120) { vx = eb1[i - 90];  vm = mb1[i - 90]; }
        else if (i < 420) { vx = ew2[i - 120]; vm = mw2[i - 120]; }
        else              { vx = eb2[i - 420]; vm = mb2[i - 420]; }
        sW[i] = vx; sW[430 + i] = vm;
    }
    __syncthreads();
    int idx = blockIdx.x * blockDim.x + tid;
    if (idx >= n) return;
    float a1x = ea1[0], a2x = ea2[0], a1m = ma1[0], a2m = ma2[0];
    float xi[3], mi[3];
    for (int c = 0; c < 3; c++) { xi[c] = x[idx * 3 + c]; mi[c] = mask[idx * 3 + c]; }
    float hx[30], hm[30];
#pragma unroll
    for (int j = 0; j < 30; j++) {
        float sx = sW[90 + j], sm = sW[430 + 90 + j];
        for (int c = 0; c < 3; c++) { sx += xi[c] * sW[c * 30 + j]; sm += mi[c] * sW[430 + c * 30 + j]; }
        hx[j] = preluf(sx, a1x); hm[j] = preluf(sm, a1m);
    }
#pragma unroll
    for (int j = 0; j < 10; j++) {
        float sx = sW[420 + j], sm = sW[430 + 420 + j];
        for (int k = 0; k < 30; k++) { sx += hx[k] * sW[120 + k * 10 + j]; sm += hm[k] * sW[430 + 120 + k * 10 + j]; }
        xe[idx * 10 + j] = preluf(sx, a2x);
        me[idx * 10 + j] = preluf(sm, a2m);
    }
}

// ---------------------------------------------------------------- per-layer global feature: g = prelu(xin@fg+b, a2); segment-sum via LDS reduction
__global__ void global_feat_kernel(const float* __restrict__ xin, int cin,
                                   const int* __restrict__ batch,
                                   const float* __restrict__ fgw, const float* __restrict__ fgb,
                                   const float* __restrict__ aArr,
                                   float* __restrict__ gpsum, float* __restrict__ gpcnt, int n) {
    __shared__ float sSum[NB * 3];
    __shared__ float sCnt[NB];
    int tid = threadIdx.x;
    if (tid < NB * 3) sSum[tid] = 0.0f;
    if (tid < NB) sCnt[tid] = 0.0f;
    __syncthreads();
    int i = blockIdx.x * blockDim.x + tid;
    if (i < n) {
        int b = batch[i];
        float a2 = aArr[2];
        for (int c = 0; c < 3; c++) {
            float s = fgb[c];
            for (int k = 0; k < cin; k++) s += xin[i * cin + k] * fgw[k * 3 + c];
            atomicAdd(&sSum[b * 3 + c], preluf(s, a2));
        }
        atomicAdd(&sCnt[b], 1.0f);
    }
    __syncthreads();
    if (tid < NB * 3) atomicAdd(&gpsum[tid], sSum[tid]);
    if (tid < NB) atomicAdd(&gpcnt[tid], sCnt[tid]);
}

__global__ void gp_div_kernel(const float* __restrict__ gpsum, const float* __restrict__ gpcnt,
                              float* __restrict__ gp) {
    int t = threadIdx.x;
    if (t < NB * 3) gp[t] = gpsum[t] / fmaxf(gpcnt[t / 3], 1.0f);
}

// ---------------------------------------------------------------- edge message kernel: fc1 over 16-edge tiles via V_WMMA_F32_16X16X4_F32
template <int CIN, int KPAD>
__global__ void edge_msg_kernel(const float* __restrict__ xin,
                                const float* __restrict__ me,
                                const float* __restrict__ pos,
                                const int* __restrict__ srcIdx, const int* __restrict__ dstIdx,
                                const int* __restrict__ batch,
                                const float* __restrict__ gp,
                                const float* __restrict__ feW, const float* __restrict__ feB,
                                const float* __restrict__ fc1W, const float* __restrict__ fc1B,
                                const float* __restrict__ aArr,
                                float* __restrict__ aggSum, float* __restrict__ aggCnt,
                                int nEdges) {
    constexpr int FIN = CIN + 28;                // [xin(CIN), me(10), gpr(3), e(15)]
    __shared__ float As[4 * 16 * ASTR];          // per-wave 16xKPAD A tiles
    __shared__ float Ws[KMAX * 16];              // fc1 weights, KPAD x 16 (col 15 zero)
    __shared__ float sFeW[45], sFeB[15], sB1[15];
    __shared__ int   sDst[4 * 16];
    int tid = threadIdx.x;
    for (int i = tid; i < KPAD * 16; i += blockDim.x) {
        int k = i >> 4, nn = i & 15;
        Ws[i] = (k < FIN && nn < 15) ? fc1W[k * 15 + nn] : 0.0f;
    }
    if (tid < 45) sFeW[tid] = feW[tid];
    if (tid < 15) { sFeB[tid] = feB[tid]; sB1[tid] = fc1B[tid]; }
    __syncthreads();

    int lane = tid & 31, wave = tid >> 5;
    float* A = &As[wave * 16 * ASTR];
    int* dl = &sDst[wave * 16];
    int e0 = (blockIdx.x * 4 + wave) * 16;
    float a3 = aArr[3], a0 = aArr[0];

    if (lane < 16) {                             // gather one edge per lane into the A tile
        int e = e0 + lane;
        bool valid = e < nEdges;
        int s = 0, d = 0;
        if (valid) { s = srcIdx[e]; d = dstIdx[e]; }
        dl[lane] = valid ? d : -1;
        float rx = 0.f, ry = 0.f, rz = 0.f;
        if (valid) {
            rx = pos[d * 3 + 0] - pos[s * 3 + 0];
            ry = pos[d * 3 + 1] - pos[s * 3 + 1];
            rz = pos[d * 3 + 2] - pos[s * 3 + 2];
        }
        float* Ar = &A[lane * ASTR];
        int k = 0;
        for (int c = 0; c < CIN; c++) Ar[k++] = valid ? xin[s * CIN + c] : 0.0f;
        for (int c = 0; c < 10; c++)  Ar[k++] = valid ? me[s * 10 + c] : 0.0f;
        int b = valid ? batch[s] : 0;
        for (int c = 0; c < 3; c++)   Ar[k++] = valid ? gp[b * 3 + c] : 0.0f;
#pragma unroll
        for (int nn = 0; nn < 15; nn++) {
            float v = rx * sFeW[0 * 15 + nn] + ry * sFeW[1 * 15 + nn] + rz * sFeW[2 * 15 + nn] + sFeB[nn];
            Ar[k++] = valid ? preluf(v, a3) : 0.0f;
        }
        for (; k < KPAD; k++) Ar[k] = 0.0f;
        if (valid) atomicAdd(&aggCnt[d], 1.0f);  // in-degree for the segment mean
    }
    __syncthreads();                             // all 32 lanes converged: EXEC all-ones for WMMA

    v8f acc = {};
    int row = lane & 15;
    int hi = (lane >> 4) << 1;                   // lanes 0-15: K=k,k+1 ; lanes 16-31: K=k+2,k+3
#pragma unroll
    for (int kk = 0; kk < KPAD; kk += 4) {
        int k0 = kk + hi;
        v2f av; av.x = A[row * ASTR + k0];  av.y = A[row * ASTR + k0 + 1];
        v2f bv; bv.x = Ws[k0 * 16 + row];   bv.y = Ws[(k0 + 1) * 16 + row];
        acc = __builtin_amdgcn_wmma_f32_16x16x4_f32(false, av, false, bv, (short)0, acc, false, false);
    }

    // D layout: lane n=lane&15; VGPR r -> edge m = r + (lane<16 ? 0 : 8)
    int nfeat = row;
    if (nfeat < 15) {
        float bias = sB1[nfeat];
        int mofs = (lane < 16) ? 0 : 8;
#pragma unroll
        for (int r = 0; r < 8; r++) {
            int d = dl[r + mofs];
            if (d >= 0) {
                float v = preluf(acc[r] + bias, a0);
                atomicAdd(&aggSum[d * 15 + nfeat], v);
            }
        }
    }
}

// ---------------------------------------------------------------- node fc2: out = prelu(cat(xin,me,agg,gpr)@W2+b, a1) (+residual)
template <int CIN>
__global__ void node_fc2_kernel(const float* __restrict__ xin,
                                const float* __restrict__ me,
                                const float* __restrict__ aggSum, const float* __restrict__ aggCnt,
                                const int* __restrict__ batch,
                                const float* __restrict__ gp,
                                const float* __restrict__ fc2W, const float* __restrict__ fc2B,
                                const float* __restrict__ aArr,
                                const float* __restrict__ resid,
                                float* __restrict__ out, int n) {
    constexpr int FIN2 = CIN + 28;
    __shared__ float W[KMAX * 20];
    __shared__ float Bv[20];
    int tid = threadIdx.x;
    for (int i = tid; i < FIN2 * 20; i += blockDim.x) W[i] = fc2W[i];
    if (tid < 20) Bv[tid] = fc2B[tid];
    __syncthreads();
    int i = blockIdx.x * blockDim.x + tid;
    if (i >= n) return;
    float f[FIN2];
    int k = 0;
    for (int c = 0; c < CIN; c++) f[k++] = xin[i * CIN + c];
    for (int c = 0; c < 10; c++)  f[k++] = me[i * 10 + c];
    float invc = 1.0f / fmaxf(aggCnt[i], 1.0f);
    for (int c = 0; c < 15; c++)  f[k++] = aggSum[i * 15 + c] * invc;
    int b = batch[i];
    for (int c = 0; c < 3; c++)   f[k++] = gp[b * 3 + c];
    float a1 = aArr[1];
#pragma unroll 4
    for (int j = 0; j < 20; j++) {
        float s = Bv[j];
        for (int kk = 0; kk < FIN2; kk++) s += f[kk] * W[kk * 20 + j];
        s = preluf(s, a1);
        if (resid) s += resid[i * 20 + j];
        out[i * 20 + j] = s;
    }
}

// ---------------------------------------------------------------- final pooling (20 channels) + head
__global__ void pool_kernel(const float* __restrict__ xin, const int* __restrict__ batch,
                            float* __restrict__ poolSum, float* __restrict__ poolCnt, int n) {
    __shared__ float sS[NB * 20];
    __shared__ float sC[NB];
    int tid = threadIdx.x;
    for (int i = tid; i < NB * 20; i += blockDim.x) sS[i] = 0.0f;
    if (tid < NB) sC[tid] = 0.0f;
    __syncthreads();
    int i = blockIdx.x * blockDim.x + tid;
    if (i < n) {
        int b = batch[i];
        for (int c = 0; c < 20; c++) atomicAdd(&sS[b * 20 + c], xin[i * 20 + c]);
        atomicAdd(&sC[b], 1.0f);
    }
    __syncthreads();
    for (int i = tid; i < NB * 20; i += blockDim.x) atomicAdd(&poolSum[i], sS[i]);
    if (tid < NB) atomicAdd(&poolCnt[tid], sC[tid]);
}

__global__ void head_kernel(const float* __restrict__ poolSum, const float* __restrict__ poolCnt,
                            const float* __restrict__ pw1, const float* __restrict__ pb1,
                            const float* __restrict__ pa,
                            const float* __restrict__ pw2, const float* __restrict__ pb2,
                            float* __restrict__ out) {
    __shared__ float gp[NB * 20];
    __shared__ float h[NB * 20];
    int tid = threadIdx.x;
    if (tid < NB * 20) gp[tid] = poolSum[tid] / fmaxf(poolCnt[tid / 20], 1.0f);
    __syncthreads();
    if (tid < NB * 20) {
        int b = tid / 20, j = tid % 20;
        float s = pb1[j];
        for (int k = 0; k < 20; k++) s += gp[b * 20 + k] * pw1[k * 20 + j];
        h[tid] = preluf(s, pa[0]);
    }
    __syncthreads();
    if (tid < NB) {
        float s = pb2[0];
        for (int j = 0; j < 20; j++) s += h[tid * 20 + j] * pw2[j];
        out[tid] = 5.0f * s;
    }
}

// ---------------------------------------------------------------- host orchestration
extern "C" void kernel_launch(void* const* d_in, const int* in_sizes, int n_in,
                              void* d_out, int out_size, void* d_ws, size_t ws_size,
                              hipStream_t stream) {
    (void)n_in; (void)out_size; (void)ws_size;
    const float* x      = (const float*)d_in[0];
    const float* mask   = (const float*)d_in[1];
    const int*   Aed    = (const int*)d_in[2];
    const float* pos    = (const float*)d_in[3];
    const int*   batch  = (const int*)d_in[4];
    /* d_in[5] = n_nodes (unused; repeat == gp[batch[i]]) */
    const float* ew1 = (const float*)d_in[6],  *eb1 = (const float*)d_in[7];
    const float* ew2 = (const float*)d_in[8],  *eb2 = (const float*)d_in[9];
    const float* ea1 = (const float*)d_in[10], *ea2 = (const float*)d_in[11];
    const float* mw1 = (const float*)d_in[12], *mb1 = (const float*)d_in[13];
    const float* mw2 = (const float*)d_in[14], *mb2 = (const float*)d_in[15];
    const float* ma1 = (const float*)d_in[16], *ma2 = (const float*)d_in[17];
    const float* l1_fe_w  = (const float*)d_in[18], *l1_fe_b  = (const float*)d_in[19];
    const float* l1_fc1_w = (const float*)d_in[20], *l1_fc1_b = (const float*)d_in[21];
    const float* l1_fc2_w = (const float*)d_in[22], *l1_fc2_b = (const float*)d_in[23];
    const float* l1_fg_w  = (const float*)d_in[24], *l1_fg_b  = (const float*)d_in[25];
    const float* l1_a     = (const float*)d_in[26];
    const float* L_fe_w  = (const float*)d_in[27], *L_fe_b  = (const float*)d_in[28];
    const float* L_fc1_w = (const float*)d_in[29], *L_fc1_b = (const float*)d_in[30];
    const float* L_fc2_w = (const float*)d_in[31], *L_fc2_b = (const float*)d_in[32];
    const float* L_fg_w  = (const float*)d_in[33], *L_fg_b  = (const float*)d_in[34];
    const float* L_a     = (const float*)d_in[35];
    const float* pw1 = (const float*)d_in[36], *pb1 = (const float*)d_in[37];
    const float* pa  = (const float*)d_in[38];
    const float* pw2 = (const float*)d_in[39], *pb2 = (const float*)d_in[40];

    const int N = in_sizes[0] / 3;
    const int E = in_sizes[2] / 2;
    const int* srcIdx = Aed;
    const int* dstIdx = Aed + E;

    // workspace carve-up (floats)
    float* ws = (float*)d_ws;
    float* b0      = ws;
    float* b1      = b0 + (size_t)N * 20;
    float* b2      = b1 + (size_t)N * 20;
    float* meb     = b2 + (size_t)N * 20;
    float* aggSum  = meb + (size_t)N * 10;
    float* aggCnt  = aggSum + (size_t)N * 15;   // contiguous after aggSum
    float* gpsum   = aggCnt + N;
    float* gpcnt   = gpsum + NB * 3;            // contiguous after gpsum
    float* gp      = gpcnt + NB;
    float* poolSum = gp + NB * 3;
    float* poolCnt = poolSum + NB * 20;         // contiguous after poolSum

    const int TPB = 256;
    const int nodeBlocks = (N + TPB - 1) / TPB;

    encoder_kernel<<<nodeBlocks, TPB, 0, stream>>>(x, mask, ew1, eb1, ew2, eb2, ea1, ea2,
                                                   mw1, mb1, mw2, mb2, ma1, ma2, b0, meb, N);

    // residual schedule: out1@L0, out2@L3, out3@L6 live in b1 when consumed
    const float* inBuf[10]  = { b0, b1, b2, b0, b1, b2, b0, b1, b2, b0 };
    float*       outBuf[10] = { b1, b2, b0, b1, b2, b0, b1, b2, b0, b1 };
    const float* resBuf[10] = { 0, 0, b1, 0, 0, b1, 0, 0, b1, 0 };

    const int edgeTiles  = (E + 15) / 16;
    const int edgeBlocks = (edgeTiles + 3) / 4;

    for (int L = 0; L < 10; L++) {
        const float *fe_w, *fe_b, *fc1_w, *fc1_b, *fc2_w, *fc2_b, *fg_w, *fg_b, *aArr;
        int cin;
        if (L == 0) {
            fe_w = l1_fe_w; fe_b = l1_fe_b; fc1_w = l1_fc1_w; fc1_b = l1_fc1_b;
            fc2_w = l1_fc2_w; fc2_b = l1_fc2_b; fg_w = l1_fg_w; fg_b = l1_fg_b;
            aArr = l1_a; cin = 10;
        } else {
            int i = L - 1;
            fe_w  = L_fe_w  + (size_t)i * 45;      fe_b  = L_fe_b  + (size_t)i * 15;
            fc1_w = L_fc1_w + (size_t)i * 48 * 15; fc1_b = L_fc1_b + (size_t)i * 15;
            fc2_w = L_fc2_w + (size_t)i * 48 * 20; fc2_b = L_fc2_b + (size_t)i * 20;
            fg_w  = L_fg_w  + (size_t)i * 60;      fg_b  = L_fg_b  + (size_t)i * 3;
            aArr  = L_a     + (size_t)i * 4;       cin = 20;
        }

        fill_zero<<<1, 64, 0, stream>>>(gpsum, NB * 4);                       // gpsum + gpcnt
        global_feat_kernel<<<nodeBlocks, TPB, 0, stream>>>(inBuf[L], cin, batch, fg_w, fg_b, aArr,
                                                           gpsum, gpcnt, N);
        gp_div_kernel<<<1, 32, 0, stream>>>(gpsum, gpcnt, gp);
        fill_zero<<<(N * 16 + TPB - 1) / TPB, TPB, 0, stream>>>(aggSum, N * 16); // aggSum + aggCnt

        if (cin == 10) {
            edge_msg_kernel<10, 40><<<edgeBlocks, 128, 0, stream>>>(
                inBuf[L], meb, pos, srcIdx, dstIdx, batch, gp,
                fe_w, fe_b, fc1_w, fc1_b, aArr, aggSum, aggCnt, E);
            node_fc2_kernel<10><<<nodeBlocks, TPB, 0, stream>>>(
                inBuf[L], meb, aggSum, aggCnt, batch, gp, fc2_w, fc2_b, aArr,
                resBuf[L], outBuf[L], N);
        } else {
            edge_msg_kernel<20, 48><<<edgeBlocks, 128, 0, stream>>>(
                inBuf[L], meb, pos, srcIdx, dstIdx, batch, gp,
                fe_w, fe_b, fc1_w, fc1_b, aArr, aggSum, aggCnt, E);
            node_fc2_kernel<20><<<nodeBlocks, TPB, 0, stream>>>(
                inBuf[L], meb, aggSum, aggCnt, batch, gp, fc2_w, fc2_b, aArr,
                resBuf[L], outBuf[L], N);
        }
    }

    fill_zero<<<1, TPB, 0, stream>>>(poolSum, NB * 21);                       // poolSum + poolCnt
    pool_kernel<<<nodeBlocks, TPB, 0, stream>>>(b1, batch, poolSum, poolCnt, N);
    head_kernel<<<1, TPB, 0, stream>>>(poolSum, poolCnt, pw1, pb1, pa, pw2, pb2, (float*)d_out);
}